// FCOSLayer_22840636080477
// MI455X (gfx1250) — compile-verified
//
#include <hip/hip_runtime.h>
#include <cstdint>
#include <cstddef>

// Problem constants (fixed by the reference setup)
#define NB    16
#define NK    50
#define NA    3
#define NG    64
#define NCLS  80
#define NCH   85          // 5 + NCLS
#define NANCH 9
#define PLANE (NG*NG)     // 4096
#define CELLS (NB*NA*PLANE) // 196608
#define FEPS  1e-16f

struct Entry { int cell; int tcls; float t[4]; };

typedef __attribute__((ext_vector_type(2))) float v2f;
typedef __attribute__((ext_vector_type(8))) float v8f;

#if __has_builtin(__builtin_amdgcn_global_load_async_to_lds_b32) && \
    __has_builtin(__builtin_amdgcn_s_wait_asynccnt)
#define USE_ASYNC_LDS 1
typedef __attribute__((address_space(1))) int as1_int;
typedef __attribute__((address_space(3))) int as3_int;
#else
#define USE_ASYNC_LDS 0
#endif

__device__ __forceinline__ float softplusf(float x) {
  // log(1 + e^x), numerically stable
  return fmaxf(x, 0.0f) + log1pf(expf(-fabsf(x)));
}

// ---------------------------------------------------------------------------
// Kernel 0: zero gt_mask (as u32 words) and the output accumulator
// ---------------------------------------------------------------------------
__global__ void zero_kernel(unsigned int* __restrict__ mask32,
                            float* __restrict__ out) {
  int i = blockIdx.x * blockDim.x + threadIdx.x;
  if (i < CELLS / 4) mask32[i] = 0u;
  if (i == 0) *out = 0.0f;
}

// ---------------------------------------------------------------------------
// Kernel 1: per-batch label prep (16 lanes, sequential over k for XLA
// scatter "last write wins" semantics). Emits:
//  - dense gt_mask
//  - deduped sparse target list (cell, tcls, tgt_ltrb[4]) per batch
//  - gtbox table [NB][NK][5] = {x0,y0,x1,y1, 0.6*areaB}, zeroed when invalid
//    (zero box => inter=0 => ignore-condition false, same as reference's -1)
// ---------------------------------------------------------------------------
__global__ void prep_kernel(const float* __restrict__ labels,
                            const float* __restrict__ anchors,
                            const int*   __restrict__ img,
                            uint8_t* __restrict__ mask,
                            int*     __restrict__ counts,
                            Entry*   __restrict__ entries,
                            float*   __restrict__ gtbox) {
  int b = threadIdx.x;
  if (b >= NB) return;
  float imgf = (float)img[0];
  Entry* eb = entries + b * NK;
  float* gb = gtbox + b * NK * 5;
  int cnt = 0;
  for (int k = 0; k < NK; ++k) {
    const float* lab = labels + (b * NK + k) * 5;
    float l0 = lab[0], l1 = lab[1], l2 = lab[2], l3 = lab[3], l4 = lab[4];
    bool valid_row = (l0 + l1 + l2 + l3 + l4) > 0.0f;

    // gt box table for the ignore-mask pass (zero box when row invalid)
    float bx0 = 0.f, by0 = 0.f, bx1 = 0.f, by1 = 0.f, cB = 0.f;
    if (valid_row) {
      bx0 = l1 - 0.5f * l3; by0 = l2 - 0.5f * l4;
      bx1 = l1 + 0.5f * l3; by1 = l2 + 0.5f * l4;
      cB  = 0.6f * (l3 * l4);
    }
    gb[5*k+0] = bx0; gb[5*k+1] = by0; gb[5*k+2] = bx1; gb[5*k+3] = by1;
    gb[5*k+4] = cB;

    // best anchor over all 9 (normalized by img_size)
    float tw = l3, th = l4;
    int best = 0; float bestv = -1.0f;
    for (int j = 0; j < NANCH; ++j) {
      float aw = anchors[2 * j] / imgf, ah = anchors[2 * j + 1] / imgf;
      float inter = fminf(tw, aw) * fminf(th, ah);
      float uni = tw * th + aw * ah - inter;
      float iou = inter / (uni + FEPS);
      if (iou > bestv) { bestv = iou; best = j; }   // strict '>' == argmax first
    }
    int best_n = best % NA;
    bool valid = valid_row && (best < NA);          // ANCHOR_INDICES = {0,1,2}
    if (!valid) continue;

    // target ltrb in log space
    float cx = l1 * NG, cy = l2 * NG, w = l3 * NG, h = l4 * NG;
    float xc = floorf(cx) + 0.5f, yc = floorf(cy) + 0.5f;
    float lt[4];
    lt[0] = xc - (cx - 0.5f * w);
    lt[1] = yc - (cy - 0.5f * h);
    lt[2] = cx + 0.5f * w - xc;
    lt[3] = cy + 0.5f * h - yc;
    float anw = anchors[2 * best_n] / imgf;
    float tg[4];
    for (int i = 0; i < 4; ++i) {
      float v = fmaxf(lt[i], 0.0f) * (1.0f / NG);
      tg[i] = logf(v / anw + FEPS);
    }
    int ti = (int)cx, tj = (int)cy;
    int cell = ((b * NA + best_n) * NG + tj) * NG + ti;
    mask[cell] = 1;
    // dedupe: last label mapping to a cell wins
    int idx = -1;
    for (int j = 0; j < cnt; ++j) if (eb[j].cell == cell) { idx = j; break; }
    if (idx < 0) idx = cnt++;
    eb[idx].cell = cell; eb[idx].tcls = (int)l0;
    for (int i = 0; i < 4; ++i) eb[idx].t[i] = tg[i];
  }
  counts[b] = cnt;
}

// ---------------------------------------------------------------------------
// Kernel 2: fused per-cell pass over channels 0..4 only (3.9 MB of the
// 67 MB input). Decode box, division-free ignore test vs 50 LDS-resident
// gt boxes, conf BCE. gt table staged to LDS via CDNA5 async-to-LDS loads
// (ASYNCcnt) when the toolchain exposes the builtins.
// Wave32 reduction via V_WMMA_F32_16X16X4_F32 (A = lane partials, B = ones:
// D row m = p[m]+p[m+16]; per-lane sum of 8 acc VGPRs + shfl_xor(16)).
// ---------------------------------------------------------------------------
__global__ void conf_kernel(const float* __restrict__ raw,
                            const float* __restrict__ gtbox,
                            const float* __restrict__ anchors,
                            const int*   __restrict__ img,
                            const uint8_t* __restrict__ mask,
                            const int*   __restrict__ counts,
                            float* __restrict__ out) {
  __shared__ float slab[NK * 5];
  __shared__ float wred[8];

  int cell  = blockIdx.x * blockDim.x + threadIdx.x;   // grid == CELLS exactly
  int gx    = cell & (NG - 1);
  int gy    = (cell >> 6) & (NG - 1);
  int plane = cell >> 12;            // b*3 + a (block-uniform: 256 | 4096)
  int a     = plane % NA;
  int b     = plane / NA;

  // stage this batch's 250-dword gt-box table into LDS
#if USE_ASYNC_LDS
  if (threadIdx.x < NK * 5) {
    __builtin_amdgcn_global_load_async_to_lds_b32(
        (as1_int*)(gtbox + b * NK * 5 + threadIdx.x),
        (as3_int*)&slab[threadIdx.x], 0, 0);
  }
  __builtin_amdgcn_s_wait_asynccnt(0);
#else
  if (threadIdx.x < NK * 5)
    slab[threadIdx.x] = gtbox[b * NK * 5 + threadIdx.x];
#endif
  __syncthreads();

  float imgf = (float)img[0];
  float aw   = anchors[2 * a];                          // anchor width (pixels)
  size_t base = (size_t)plane * NCH * PLANE + (size_t)gy * NG + gx;

  float rl = raw[base + 0 * PLANE];
  float rt = raw[base + 1 * PLANE];
  float rr = raw[base + 2 * PLANE];
  float rb = raw[base + 3 * PLANE];
  float cl = raw[base + 4 * PLANE];                     // conf logit

  // decode: exp(raw)*anch_w / STRIDE -> grid units; *STRIDE/img -> normalized
  float k8 = aw * 0.125f;
  float el = expf(rl) * k8, et = expf(rt) * k8;
  float er = expf(rr) * k8, ebv = expf(rb) * k8;
  float s   = 8.0f / imgf;
  float pcx = ((float)gx + 0.5f + 0.5f * (er - el)) * s;
  float pcy = ((float)gy + 0.5f + 0.5f * (ebv - et)) * s;
  float pw  = (el + er) * s;
  float ph  = (et + ebv) * s;
  float ax0 = pcx - 0.5f * pw, ay0 = pcy - 0.5f * ph;
  float ax1 = pcx + 0.5f * pw, ay1 = pcy + 0.5f * ph;
  // iou > 0.6  <=>  1.6*I > 0.6*(A + B + eps)   (denominator is positive)
  float cA = 0.6f * (pw * ph + FEPS);

  bool ignore = false;
  for (int k = 0; k < NK; ++k) {
    float bx0 = slab[5*k+0], by0 = slab[5*k+1];
    float bx1 = slab[5*k+2], by1 = slab[5*k+3];
    float cB  = slab[5*k+4];
    float wx = fmaxf(fminf(ax1, bx1) - fmaxf(ax0, bx0), 0.0f);
    float wy = fmaxf(fminf(ay1, by1) - fmaxf(ay0, by0), 0.0f);
    ignore = ignore || (1.6f * (wx * wy) > cA + cB);
  }
  bool m         = mask[cell] != 0;
  bool has_valid = counts[b] > 0;
  bool cmask     = (has_valid ? !ignore : true) || m;
  float g   = m ? 1.0f : 0.0f;
  float bce = softplusf(cl) - g * cl;                   // g*sp(-x)+(1-g)*sp(x)
  float partial = cmask ? bce : 0.0f;

  // --- wave32 reduction via WMMA (EXEC all ones here: no divergence) ---
  float wsum;
#if __has_builtin(__builtin_amdgcn_wmma_f32_16x16x4_f32)
  {
    v2f aM; aM.x = partial; aM.y = 0.0f;                // A: 16x4, K=1,3 zeroed
    v2f bM; bM.x = 1.0f;    bM.y = 1.0f;                // B: ones
    v8f cM = {0.f,0.f,0.f,0.f,0.f,0.f,0.f,0.f};
    cM = __builtin_amdgcn_wmma_f32_16x16x4_f32(false, aM, false, bM,
                                               (short)0, cM, false, false);
    float srow = ((cM[0]+cM[1])+(cM[2]+cM[3])) + ((cM[4]+cM[5])+(cM[6]+cM[7]));
    wsum = srow + __shfl_xor(srow, 16, 32);             // rows 0-7 + rows 8-15
  }
#else
  {
    float p = partial;
    for (int off = 16; off; off >>= 1) p += __shfl_xor(p, off, 32);
    wsum = p;
  }
#endif
  int lane = threadIdx.x & 31, wid = threadIdx.x >> 5;
  if (lane == 0) wred[wid] = wsum;
  __syncthreads();
  if (threadIdx.x == 0) {
    float tsum = 0.0f;
    for (int i = 0; i < 8; ++i) tsum += wred[i];
    atomicAdd(out, tsum);
  }
}

// ---------------------------------------------------------------------------
// Kernel 3: sparse pass — only the <=800 gt cells touch the 80 class
// channels (gathers ~256KB instead of streaming 63MB). Also adds the
// bbox L2 term for those cells. One wave32 per entry.
// ---------------------------------------------------------------------------
__global__ void cls_kernel(const float* __restrict__ raw,
                           const int*   __restrict__ counts,
                           const Entry* __restrict__ entries,
                           float* __restrict__ out) {
  int b = blockIdx.x / NK;
  int i = blockIdx.x % NK;
  if (i >= counts[b]) return;                            // block-uniform exit
  Entry e = entries[b * NK + i];
  int cell = e.cell;
  int gx = cell & (NG - 1), gy = (cell >> 6) & (NG - 1);
  int plane = cell >> 12;
  const float* base = raw + (size_t)plane * NCH * PLANE + (size_t)gy * NG + gx;
  int lane = threadIdx.x;
  float p = 0.0f;
  // loss_cls: sum_c softplus(x_c) - x_tcls  (one-hot BCE)
  for (int c = lane; c < NCLS; c += 32)
    p += softplusf(base[(size_t)(5 + c) * PLANE]);
  if (lane == 0)
    p -= base[(size_t)(5 + e.tcls) * PLANE];
  // loss_bbox: (ltrb_raw - tgt)^2
  if (lane < 4) {
    float x = base[(size_t)lane * PLANE];
    float d = x - e.t[lane];
    p += d * d;
  }
  for (int off = 16; off; off >>= 1) p += __shfl_xor(p, off, 32);
  if (lane == 0) atomicAdd(out, p);
}

// ---------------------------------------------------------------------------
extern "C" void kernel_launch(void* const* d_in, const int* in_sizes, int n_in,
                              void* d_out, int out_size, void* d_ws, size_t ws_size,
                              hipStream_t stream) {
  (void)in_sizes; (void)n_in; (void)out_size; (void)ws_size;
  const float* raw     = (const float*)d_in[0];
  const float* labels  = (const float*)d_in[1];
  const float* anchors = (const float*)d_in[2];
  const int*   img     = (const int*)d_in[3];
  float* out = (float*)d_out;

  // workspace: [gt_mask u8 x CELLS][counts i32 x16 (64B)][entries][gtbox]
  uint8_t* mask    = (uint8_t*)d_ws;
  int*     counts  = (int*)((char*)d_ws + CELLS);
  Entry*   entries = (Entry*)((char*)d_ws + CELLS + 64);
  float*   gtbox   = (float*)((char*)d_ws + CELLS + 64 + NB * NK * (int)sizeof(Entry));

  zero_kernel<<<CELLS / (4 * 256), 256, 0, stream>>>((unsigned int*)mask, out);
  prep_kernel<<<1, 32, 0, stream>>>(labels, anchors, img, mask, counts,
                                    entries, gtbox);
  conf_kernel<<<CELLS / 256, 256, 0, stream>>>(raw, gtbox, anchors, img,
                                               mask, counts, out);
  cls_kernel<<<NB * NK, 32, 0, stream>>>(raw, counts, entries, out);
}